// Sinkhorn_87007447482780
// MI455X (gfx1250) — compile-verified
//
#include <hip/hip_runtime.h>
#include <math.h>

typedef float v2f __attribute__((ext_vector_type(2)));
typedef float v8f __attribute__((ext_vector_type(8)));

#define N0       2048                 // real block size
#define NPAD     4096                 // padded square size
#define EPS      0.05f
#define INV_EPS  20.0f                // 1/EPS
#define STEPS    50
#define LOG_NPAD 8.31776616671934f    // log(4096)

#define TILES    (N0 / 16)            // 128 row/col tiles
#define WPB      8                    // waves per block (256 threads)
#define KSLICE   (N0 / WPB)           // 256 reduction columns per wave

// ------------------------------------------------------------------
// init v = 0
__global__ void sk_init(float* __restrict__ v) {
    int i = blockIdx.x * blockDim.x + threadIdx.x;
    if (i < NPAD) v[i] = 0.0f;
}

// ------------------------------------------------------------------
// one-block reduce over a 4096-vector x:
//   out3[0] = g    = max(x)
//   out3[1] = Hpad = sum_{j>=2048} exp((x_j-g)/eps)
//   out3[2] = Hall = sum_{all j}  exp((x_j-g)/eps)
__global__ __launch_bounds__(1024) void sk_reduce(const float* __restrict__ x,
                                                  float* __restrict__ out3) {
    __shared__ float sm[1024];
    int t = threadIdx.x;
    float a0 = x[t], a1 = x[t + 1024], a2 = x[t + 2048], a3 = x[t + 3072];
    float mx = fmaxf(fmaxf(a0, a1), fmaxf(a2, a3));
    sm[t] = mx; __syncthreads();
    for (int s = 512; s > 0; s >>= 1) {
        if (t < s) sm[t] = fmaxf(sm[t], sm[t + s]);
        __syncthreads();
    }
    float g = sm[0]; __syncthreads();

    float e0 = __expf((a0 - g) * INV_EPS);
    float e1 = __expf((a1 - g) * INV_EPS);
    float e2 = __expf((a2 - g) * INV_EPS);   // index >= 2048
    float e3 = __expf((a3 - g) * INV_EPS);   // index >= 2048

    sm[t] = e0 + e1 + e2 + e3; __syncthreads();
    for (int s = 512; s > 0; s >>= 1) {
        if (t < s) sm[t] += sm[t + s];
        __syncthreads();
    }
    float hall = sm[0]; __syncthreads();

    sm[t] = e2 + e3; __syncthreads();
    for (int s = 512; s > 0; s >>= 1) {
        if (t < s) sm[t] += sm[t + s];
        __syncthreads();
    }
    if (t == 0) { out3[0] = g; out3[1] = sm[0]; out3[2] = hall; }
}

// ------------------------------------------------------------------
// Row half-step: u_new_i = eps*logp - g - eps*log(S_i + Hpad)
//   S_i = sum_{j<2048} exp((v_j - C_ij - g)/eps)
// 256-thread blocks (8 wave32s). Each wave WMMA-reduces a 256-column
// K-slice of a 16-row tile (V_WMMA_F32_16X16X4_F32, B = ones => every
// D column is the row-sum; no shuffles). Two independent accumulator
// chains for matrix-pipe ILP; LDS combine across waves (deterministic).
// Blocks >= TILES fill the closed-form padded rows.
__global__ __launch_bounds__(256) void sk_row(const float* __restrict__ C,
                                              const float* __restrict__ v,
                                              float* __restrict__ u,
                                              const float* __restrict__ scal) {
    __shared__ float part[WPB * 16];
    const float g = scal[0], Hpad = scal[1], Hall = scal[2];
    const float UCONST = -EPS * LOG_NPAD;   // eps * logp
    int blk = blockIdx.x;
    int tid = threadIdx.x;

    if (blk < TILES) {
        int wave = tid >> 5;
        int lane = tid & 31;
        int m  = lane & 15;     // A-matrix M (row of tile)
        int kh = lane >> 4;     // K half-select
        int rowBase = blk * 16;
        const float* row = C + (size_t)(rowBase + m) * N0;
        int base = wave * KSLICE;

        v8f acc0 = {}, acc1 = {};
        v2f ones; ones.x = 1.0f; ones.y = 1.0f;

        for (int c = base; c < base + KSLICE; c += 8) {
            int j0 = c + 2 * kh;
            int j1 = c + 4 + 2 * kh;
            float c0 = row[j0], c1 = row[j0 + 1];
            float c2 = row[j1], c3 = row[j1 + 1];
            float w0 = v[j0], w1 = v[j0 + 1];
            float w2 = v[j1], w3 = v[j1 + 1];
            v2f a0, a1;
            a0.x = __expf((w0 - c0 - g) * INV_EPS);
            a0.y = __expf((w1 - c1 - g) * INV_EPS);
            a1.x = __expf((w2 - c2 - g) * INV_EPS);
            a1.y = __expf((w3 - c3 - g) * INV_EPS);
            acc0 = __builtin_amdgcn_wmma_f32_16x16x4_f32(
                       false, a0, false, ones, (short)0, acc0, false, false);
            acc1 = __builtin_amdgcn_wmma_f32_16x16x4_f32(
                       false, a1, false, ones, (short)0, acc1, false, false);
        }
        v8f acc = acc0 + acc1;
        // D layout: lanes 0-15 hold M=0..7 in acc[0..7]; lanes 16-31 hold M=8..15
        float S = 0.0f; int mo = -1;
        if (lane < 8)                     { S = acc[lane];      mo = lane;     }
        else if (lane >= 16 && lane < 24) { S = acc[lane - 16]; mo = lane - 8; }
        if (mo >= 0) part[wave * 16 + mo] = S;
        __syncthreads();
        if (tid < 16) {
            float Ssum = 0.0f;
            #pragma unroll
            for (int w = 0; w < WPB; ++w) Ssum += part[w * 16 + tid];
            u[rowBase + tid] = UCONST - g - EPS * __logf(Ssum + Hpad);
        }
    } else {
        int idx = (blk - TILES) * 256 + tid;
        if (idx < N0)
            u[N0 + idx] = UCONST - g - EPS * __logf(Hall);
    }
}

// ------------------------------------------------------------------
// Column half-step (symmetric; K runs over rows, so lanes 0-15 read 16
// consecutive floats of a row slice -> coalesced 64B segments).
__global__ __launch_bounds__(256) void sk_col(const float* __restrict__ C,
                                              const float* __restrict__ u,
                                              float* __restrict__ vout,
                                              const float* __restrict__ scal) {
    __shared__ float part[WPB * 16];
    const float g = scal[0], Hpad = scal[1], Hall = scal[2];
    const float VCONST = -EPS * LOG_NPAD;   // eps * logq
    int blk = blockIdx.x;
    int tid = threadIdx.x;

    if (blk < TILES) {
        int wave = tid >> 5;
        int lane = tid & 31;
        int m  = lane & 15;          // M = tile column
        int kh = lane >> 4;
        int colBase = blk * 16;
        int col = colBase + m;
        int base = wave * KSLICE;

        v8f acc0 = {}, acc1 = {};
        v2f ones; ones.x = 1.0f; ones.y = 1.0f;

        for (int r = base; r < base + KSLICE; r += 8) {
            int i0 = r + 2 * kh;
            int i1 = r + 4 + 2 * kh;
            float c0 = C[(size_t)i0 * N0 + col];
            float c1 = C[(size_t)(i0 + 1) * N0 + col];
            float c2 = C[(size_t)i1 * N0 + col];
            float c3 = C[(size_t)(i1 + 1) * N0 + col];
            float w0 = u[i0], w1 = u[i0 + 1];
            float w2 = u[i1], w3 = u[i1 + 1];
            v2f a0, a1;
            a0.x = __expf((w0 - c0 - g) * INV_EPS);
            a0.y = __expf((w1 - c1 - g) * INV_EPS);
            a1.x = __expf((w2 - c2 - g) * INV_EPS);
            a1.y = __expf((w3 - c3 - g) * INV_EPS);
            acc0 = __builtin_amdgcn_wmma_f32_16x16x4_f32(
                       false, a0, false, ones, (short)0, acc0, false, false);
            acc1 = __builtin_amdgcn_wmma_f32_16x16x4_f32(
                       false, a1, false, ones, (short)0, acc1, false, false);
        }
        v8f acc = acc0 + acc1;
        float S = 0.0f; int mo = -1;
        if (lane < 8)                     { S = acc[lane];      mo = lane;     }
        else if (lane >= 16 && lane < 24) { S = acc[lane - 16]; mo = lane - 8; }
        if (mo >= 0) part[wave * 16 + mo] = S;
        __syncthreads();
        if (tid < 16) {
            float Ssum = 0.0f;
            #pragma unroll
            for (int w = 0; w < WPB; ++w) Ssum += part[w * 16 + tid];
            vout[colBase + tid] = VCONST - g - EPS * __logf(Ssum + Hpad);
        }
    } else {
        int idx = (blk - TILES) * 256 + tid;
        if (idx < N0)
            vout[N0 + idx] = VCONST - g - EPS * __logf(Hall);
    }
}

// ------------------------------------------------------------------
// out[i,j] = 4096 * exp(min((u_i + v_j - C_ij)/eps, 0)), C=0 in padding
__global__ void sk_final(const float* __restrict__ C,
                         const float* __restrict__ u,
                         const float* __restrict__ v,
                         float* __restrict__ out) {
    int j = blockIdx.x * blockDim.x + threadIdx.x;
    int i = blockIdx.y;
    float c = (i < N0 && j < N0) ? C[(size_t)i * N0 + j] : 0.0f;
    float t = (u[i] + v[j] - c) * INV_EPS;
    t = fminf(t, 0.0f);
    out[(size_t)i * NPAD + j] = 4096.0f * __expf(t);
}

// ------------------------------------------------------------------
extern "C" void kernel_launch(void* const* d_in, const int* in_sizes, int n_in,
                              void* d_out, int out_size, void* d_ws, size_t ws_size,
                              hipStream_t stream) {
    (void)in_sizes; (void)n_in; (void)out_size; (void)ws_size;
    const float* C = (const float*)d_in[0];
    float* ws   = (float*)d_ws;
    float* v    = ws;                  // 4096
    float* u    = ws + NPAD;           // 4096
    float* sclV = ws + 2 * NPAD;       // 3 scalars (g, Hpad, Hall) for v
    float* sclU = ws + 2 * NPAD + 4;   // 3 scalars for u
    float* out  = (float*)d_out;

    sk_init<<<NPAD / 256, 256, 0, stream>>>(v);

    const int gridUpd = TILES + N0 / 256;   // 128 WMMA tile blocks + 8 padding blocks
    for (int s = 0; s < STEPS; ++s) {
        sk_reduce<<<1, 1024, 0, stream>>>(v, sclV);
        sk_row   <<<gridUpd, 256, 0, stream>>>(C, v, u, sclV);
        sk_reduce<<<1, 1024, 0, stream>>>(u, sclU);
        sk_col   <<<gridUpd, 256, 0, stream>>>(C, u, v, sclU);
    }

    dim3 g(NPAD / 256, NPAD);
    sk_final<<<g, 256, 0, stream>>>(C, u, v, out);
}